// SimpleGCN_55886114456064
// MI455X (gfx1250) — compile-verified
//
#include <hip/hip_runtime.h>
#include <hip/hip_bf16.h>
#include <math.h>

// ---------------------------------------------------------------------------
// GATv2 GNN forward for MI455X (gfx1250). fp32 WMMA (V_WMMA_F32_16X16X4_F32)
// for all dense GEMMs (register-blocked 16x64 per wave, fully unrolled K),
// wave32 shuffles + atomics for the graph ops, global_prefetch for B stream.
// ---------------------------------------------------------------------------

#define FEAT    32
#define HID     64
#define HEADS   6
#define HC      (HEADS * HID)   // 384
#define NLAYERS 4
#define NGRAPHS 16
#define EPS_BN  1e-5f
#define EPS_IN  1e-5f

typedef __attribute__((ext_vector_type(2))) float v2f;
typedef __attribute__((ext_vector_type(8))) float v8f;

// --- monotone float <-> unsigned key for atomicMax-based segment max --------
__device__ __forceinline__ unsigned fkey(float f) {
    unsigned b = __float_as_uint(f);
    return b ^ ((b >> 31) ? 0xFFFFFFFFu : 0x80000000u);
}
__device__ __forceinline__ float keyf(unsigned k) {
    unsigned b = k ^ ((k >> 31) ? 0x80000000u : 0xFFFFFFFFu);
    return __uint_as_float(b);
}

// ---------------------------------------------------------------------------
// fp32 GEMM: C[N,M] = A[N,K] @ B[K,M] + bias[M] (optional ReLU).
// One wave computes a 16x64 strip: 4 accumulator tiles share one A fragment
// per k-step (4 v_wmma per A load). K is compile-time -> full unroll.
// Requires N % 16 == 0, M % 64 == 0.
//
// fp32 WMMA VGPR layouts (ISA 7.12.2):
//   A 16x4:  lanes 0-15 row=lane   : v0=K0, v1=K1
//            lanes16-31 row=lane-16: v0=K2, v1=K3
//   B 4x16 (row striped across lanes, mirrored):
//            lanes 0-15 col=lane   : v0=K0 row, v1=K1 row
//            lanes16-31 col=lane-16: v0=K2 row, v1=K3 row
//   C/D 16x16: vgpr r -> row r (lanes 0-15) / row r+8 (lanes 16-31), col lane&15
// ---------------------------------------------------------------------------
template <int K>
__global__ void gemm_wmma_f32_t(const float* __restrict__ A,
                                const float* __restrict__ B,
                                const float* __restrict__ bias,
                                float* __restrict__ C,
                                int N, int M, int do_relu)
{
    const int wavesPerBlock = blockDim.x >> 5;
    const int wave    = blockIdx.x * wavesPerBlock + (threadIdx.x >> 5);
    const int tilesMG = M >> 6;                     // 64-wide column groups
    const int tilesN  = N >> 4;
    if (wave >= tilesMG * tilesN) return;           // wave-uniform: EXEC all-1s

    const int tg   = wave % tilesMG;
    const int tn   = wave / tilesMG;
    const int lane = threadIdx.x & 31;
    const int half = lane >> 4;                     // 0: lanes 0-15, 1: 16-31
    const int l    = lane & 15;

    const int rowA = (tn << 4) + l;
    const int col0 = (tg << 6) + l;

    const float* __restrict__ Arow = A + (size_t)rowA * K;

    v8f acc0 = {}, acc1 = {}, acc2 = {}, acc3 = {};

    #pragma unroll
    for (int k = 0; k < K; k += 4) {
        const int kb = k + half * 2;                // this lane's K pair base
        const v2f a = *(const v2f*)(Arow + kb);     // contiguous -> b64 load

        const float* __restrict__ B0 = B + (size_t)kb * M;
        const float* __restrict__ B1 = B + (size_t)(kb + 1) * M;

        v2f b0, b1, b2, b3;
        b0.x = B0[col0 +  0]; b0.y = B1[col0 +  0];
        b1.x = B0[col0 + 16]; b1.y = B1[col0 + 16];
        b2.x = B0[col0 + 32]; b2.y = B1[col0 + 32];
        b3.x = B0[col0 + 48]; b3.y = B1[col0 + 48];

        if (k + 4 < K) {                            // prefetch next B k-block
            __builtin_prefetch(B + (size_t)(k + 4) * M + col0, 0, 1);
            __builtin_prefetch(B + (size_t)(k + 5) * M + col0, 0, 1);
        }

        acc0 = __builtin_amdgcn_wmma_f32_16x16x4_f32(
                   false, a, false, b0, (short)0, acc0, false, false);
        acc1 = __builtin_amdgcn_wmma_f32_16x16x4_f32(
                   false, a, false, b1, (short)0, acc1, false, false);
        acc2 = __builtin_amdgcn_wmma_f32_16x16x4_f32(
                   false, a, false, b2, (short)0, acc2, false, false);
        acc3 = __builtin_amdgcn_wmma_f32_16x16x4_f32(
                   false, a, false, b3, (short)0, acc3, false, false);
    }

    v8f* accs[4] = { &acc0, &acc1, &acc2, &acc3 };
    #pragma unroll
    for (int t = 0; t < 4; ++t) {
        const int col = col0 + 16 * t;
        const float bv = bias ? bias[col] : 0.0f;
        #pragma unroll
        for (int r = 0; r < 8; ++r) {
            const int row = (tn << 4) + r + half * 8;
            float v = (*accs[t])[r] + bv;
            if (do_relu) v = fmaxf(v, 0.0f);
            C[(size_t)row * M + col] = v;
        }
    }
}

// --- edge scalar feature: ea = relu( relu(|d| * w_e1 + b_e1) @ w_e2 + b_e2 )
__global__ void edge_feat_kernel(const float* __restrict__ pos,
                                 const int* __restrict__ src,
                                 const int* __restrict__ dst,
                                 const float* __restrict__ w_e1,
                                 const float* __restrict__ b_e1,
                                 const float* __restrict__ w_e2,
                                 const float* __restrict__ b_e2,
                                 float* __restrict__ ea, int E)
{
    int e = blockIdx.x * blockDim.x + threadIdx.x;
    if (e >= E) return;
    int s = src[e], d = dst[e];
    float dx = pos[s * 3 + 0] - pos[d * 3 + 0];
    float dy = pos[s * 3 + 1] - pos[d * 3 + 1];
    float dz = pos[s * 3 + 2] - pos[d * 3 + 2];
    float s2 = dx * dx + dy * dy + dz * dz;
    float ew = (s2 == 0.0f) ? 0.0f : sqrtf(s2);
    float acc = b_e2[0];
    #pragma unroll
    for (int j = 0; j < 32; ++j) {
        float m = fmaxf(ew * w_e1[j] + b_e1[j], 0.0f);
        acc += m * w_e2[j];
    }
    ea[e] = fmaxf(acc, 0.0f);
}

// --- zero a float region (also initializes amax keys: 0u == smallest key) ---
__global__ void zero_kernel(float* __restrict__ p, size_t n)
{
    size_t i = (size_t)blockIdx.x * blockDim.x + threadIdx.x;
    if (i < n) p[i] = 0.0f;
}

// --- pass 1: alpha[e,h] = att . leaky_relu(xl[src]+xr[dst]+ea*We), seg-max --
__global__ void attn_alpha_kernel(const float* __restrict__ xl,
                                  const float* __restrict__ xr,
                                  const float* __restrict__ ea,
                                  const float* __restrict__ We,
                                  const float* __restrict__ att,
                                  const int* __restrict__ src,
                                  const int* __restrict__ dst,
                                  float* __restrict__ alpha,
                                  unsigned* __restrict__ amax, int E)
{
    const int wave = blockIdx.x * (blockDim.x >> 5) + (threadIdx.x >> 5);
    if (wave >= E) return;
    const int lane = threadIdx.x & 31;
    const int s = src[wave], d = dst[wave];
    const float eav = ea[wave];

    #pragma unroll
    for (int h = 0; h < HEADS; ++h) {
        float acc = 0.0f;
        #pragma unroll
        for (int cc = 0; cc < 2; ++cc) {
            const int idx = h * HID + lane + cc * 32;
            float g = xl[(size_t)s * HC + idx] + xr[(size_t)d * HC + idx]
                      + eav * We[idx];
            g = (g > 0.0f) ? g : 0.2f * g;
            acc += g * att[idx];
        }
        #pragma unroll
        for (int off = 16; off > 0; off >>= 1)
            acc += __shfl_xor(acc, off, 32);
        if (lane == 0) {
            alpha[(size_t)wave * HEADS + h] = acc;
            atomicMax(&amax[d * HEADS + h], fkey(acc));
        }
    }
}

// --- pass 2: ex = exp(alpha - amax[dst]); den[dst] += ex  (ex overwrites alpha)
__global__ void attn_softmax_kernel(float* __restrict__ alpha,
                                    const unsigned* __restrict__ amax,
                                    float* __restrict__ den,
                                    const int* __restrict__ dst, int E)
{
    int i = blockIdx.x * blockDim.x + threadIdx.x;
    if (i >= E * HEADS) return;
    int e = i / HEADS, h = i - e * HEADS;
    int d = dst[e];
    float am = keyf(amax[d * HEADS + h]);
    float ex = expf(alpha[i] - am);
    alpha[i] = ex;
    atomicAdd(&den[d * HEADS + h], ex);
}

// --- pass 3: agg[dst] += xl[src] * (ex / (den[dst] + 1e-16)) ----------------
__global__ void attn_agg_kernel(const float* __restrict__ xl,
                                const float* __restrict__ ex,
                                const float* __restrict__ den,
                                const int* __restrict__ src,
                                const int* __restrict__ dst,
                                float* __restrict__ agg, int E)
{
    const int wave = blockIdx.x * (blockDim.x >> 5) + (threadIdx.x >> 5);
    if (wave >= E) return;
    const int lane = threadIdx.x & 31;
    const int s = src[wave], d = dst[wave];
    #pragma unroll
    for (int h = 0; h < HEADS; ++h) {
        const float a = ex[(size_t)wave * HEADS + h]
                        / (den[d * HEADS + h] + 1e-16f);
        #pragma unroll
        for (int cc = 0; cc < 2; ++cc) {
            const int idx = h * HID + lane + cc * 32;
            atomicAdd(&agg[(size_t)d * HC + idx],
                      xl[(size_t)s * HC + idx] * a);
        }
    }
}

// --- pass 4: ht = relu(h + mean_heads(agg) + b_gat) -------------------------
__global__ void node_update_kernel(const float* __restrict__ h,
                                   const float* __restrict__ agg,
                                   const float* __restrict__ b_gat,
                                   float* __restrict__ ht, int N)
{
    int i = blockIdx.x * blockDim.x + threadIdx.x;
    if (i >= N * HID) return;
    int n = i / HID, c = i - n * HID;
    float sum = 0.0f;
    #pragma unroll
    for (int hh = 0; hh < HEADS; ++hh)
        sum += agg[(size_t)n * HC + hh * HID + c];
    ht[i] = fmaxf(h[i] + sum * (1.0f / HEADS) + b_gat[c], 0.0f);
}

// --- per-graph instance norm: contiguous 1000-node segments, 64 channels ----
__global__ void inorm_kernel(const float* __restrict__ ht,
                             float* __restrict__ h, int NP)
{
    const int g = blockIdx.x;
    const int c = threadIdx.x;                 // 64 threads = 64 channels
    const float* base = ht + (size_t)g * NP * HID;
    float s = 0.0f, s2 = 0.0f;
    for (int i = 0; i < NP; ++i) {
        float v = base[(size_t)i * HID + c];
        s += v; s2 += v * v;
    }
    const float mean = s / NP;
    const float var  = s2 / NP - mean * mean;
    const float inv  = rsqrtf(var + EPS_IN);
    float* ob = h + (size_t)g * NP * HID;
    for (int i = 0; i < NP; ++i)
        ob[(size_t)i * HID + c] = (base[(size_t)i * HID + c] - mean) * inv;
}

// --- batch-norm eval, in place ---------------------------------------------
__global__ void bn_kernel(float* __restrict__ z,
                          const float* __restrict__ g,
                          const float* __restrict__ b,
                          const float* __restrict__ m,
                          const float* __restrict__ v, int total)
{
    int i = blockIdx.x * blockDim.x + threadIdx.x;
    if (i >= total) return;
    int c = i & (HID - 1);
    z[i] = (z[i] - m[c]) * rsqrtf(v[c] + EPS_BN) * g[c] + b[c];
}

// --- pred = bn2(h_out) @ w_pred + b_pred ------------------------------------
__global__ void pred_kernel(const float* __restrict__ hout,
                            const float* __restrict__ g2,
                            const float* __restrict__ b2,
                            const float* __restrict__ m2,
                            const float* __restrict__ v2,
                            const float* __restrict__ w_pred,
                            const float* __restrict__ b_pred,
                            float* __restrict__ pred, int N)
{
    int n = blockIdx.x * blockDim.x + threadIdx.x;
    if (n >= N) return;
    const float* hr = hout + (size_t)n * HID;
    float acc = b_pred[0];
    #pragma unroll
    for (int c = 0; c < HID; ++c) {
        float t = (hr[c] - m2[c]) * rsqrtf(v2[c] + EPS_BN) * g2[c] + b2[c];
        acc += t * w_pred[c];
    }
    pred[n] = acc;
}

// ---------------------------------------------------------------------------
// Host launcher
// ---------------------------------------------------------------------------
static inline size_t align256(size_t x) { return (x + 255) & ~(size_t)255; }

extern "C" void kernel_launch(void* const* d_in, const int* in_sizes, int n_in,
                              void* d_out, int out_size, void* d_ws, size_t ws_size,
                              hipStream_t stream)
{
    const float* x      = (const float*)d_in[0];
    const float* pos    = (const float*)d_in[1];
    const float* w_lin1 = (const float*)d_in[2];
    const float* b_lin1 = (const float*)d_in[3];
    const float* w_e1   = (const float*)d_in[4];
    const float* b_e1   = (const float*)d_in[5];
    const float* w_e2   = (const float*)d_in[6];
    const float* b_e2   = (const float*)d_in[7];
    const float* Wl     = (const float*)d_in[8];
    const float* bl     = (const float*)d_in[9];
    const float* Wr     = (const float*)d_in[10];
    const float* br     = (const float*)d_in[11];
    const float* We     = (const float*)d_in[12];
    const float* att    = (const float*)d_in[13];
    const float* b_gat  = (const float*)d_in[14];
    const float* w_emb1 = (const float*)d_in[15];
    const float* b_emb1 = (const float*)d_in[16];
    const float* bn1_g  = (const float*)d_in[17];
    const float* bn1_b  = (const float*)d_in[18];
    const float* bn1_m  = (const float*)d_in[19];
    const float* bn1_v  = (const float*)d_in[20];
    const float* w_emb2 = (const float*)d_in[21];
    const float* b_emb2 = (const float*)d_in[22];
    const float* bn2_g  = (const float*)d_in[23];
    const float* bn2_b  = (const float*)d_in[24];
    const float* bn2_m  = (const float*)d_in[25];
    const float* bn2_v  = (const float*)d_in[26];
    const float* w_pred = (const float*)d_in[27];
    const float* b_pred = (const float*)d_in[28];
    const int*   ei     = (const int*)d_in[29];

    const int N  = in_sizes[0] / FEAT;          // 16000
    const int E  = in_sizes[29] / 2;            // edge count
    const int NP = N / NGRAPHS;                 // 1000

    const int* src = ei;
    const int* dst = ei + E;

    // --- workspace carve-up -------------------------------------------------
    char* wp = (char*)d_ws;
    auto allocf = [&wp](size_t nelem) {
        float* r = (float*)wp;
        wp += align256(nelem * sizeof(float));
        return r;
    };
    float*    h     = allocf((size_t)N * HID);
    float*    ht    = allocf((size_t)N * HID);
    float*    xl    = allocf((size_t)N * HC);
    float*    xr    = allocf((size_t)N * HC);
    float*    ea    = allocf((size_t)E);
    float*    alpha = allocf((size_t)E * HEADS);      // reused as ex
    float*    stats = allocf((size_t)N * (2 * HEADS + HC)); // amax|den|agg
    unsigned* amax  = (unsigned*)stats;
    float*    den   = stats + (size_t)N * HEADS;
    float*    agg   = stats + (size_t)N * 2 * HEADS;
    float*    z     = allocf((size_t)N * HID);

    float* out_h    = (float*)d_out;            // N*HID
    float* out_pred = out_h + (size_t)N * HID;  // N

    const int TB = 256;
    auto cdiv = [](long long a, long long b) { return (int)((a + b - 1) / b); };

    // --- GEMM launch helper: 4 waves (one 16x64 strip each) per 128-thr block
    auto gemm = [&](const float* A, const float* B, const float* bias,
                    float* C, int n, int k, int m, int relu) {
        int strips = (n >> 4) * (m >> 6);
        if (k == FEAT)
            gemm_wmma_f32_t<FEAT><<<cdiv(strips, 4), 128, 0, stream>>>(
                A, B, bias, C, n, m, relu);
        else
            gemm_wmma_f32_t<HID><<<cdiv(strips, 4), 128, 0, stream>>>(
                A, B, bias, C, n, m, relu);
    };

    // 0) edge scalar features + input projection
    edge_feat_kernel<<<cdiv(E, TB), TB, 0, stream>>>(pos, src, dst, w_e1, b_e1,
                                                     w_e2, b_e2, ea, E);
    gemm(x, w_lin1, b_lin1, h, N, FEAT, HID, 0);

    // 1) GATv2 layers
    for (int i = 0; i < NLAYERS; ++i) {
        const float* Wli = Wl + (size_t)i * HID * HC;
        const float* bli = bl + (size_t)i * HC;
        const float* Wri = Wr + (size_t)i * HID * HC;
        const float* bri = br + (size_t)i * HC;
        const float* Wei = We + (size_t)i * HC;
        const float* ati = att + (size_t)i * HC;
        const float* bgi = b_gat + (size_t)i * HID;

        zero_kernel<<<cdiv((size_t)N * (2 * HEADS + HC), TB), TB, 0, stream>>>(
            stats, (size_t)N * (2 * HEADS + HC));

        gemm(h, Wli, bli, xl, N, HID, HC, 0);
        gemm(h, Wri, bri, xr, N, HID, HC, 0);

        attn_alpha_kernel<<<cdiv((size_t)E * 32, TB), TB, 0, stream>>>(
            xl, xr, ea, Wei, ati, src, dst, alpha, amax, E);
        attn_softmax_kernel<<<cdiv((size_t)E * HEADS, TB), TB, 0, stream>>>(
            alpha, amax, den, dst, E);
        attn_agg_kernel<<<cdiv((size_t)E * 32, TB), TB, 0, stream>>>(
            xl, alpha, den, src, dst, agg, E);
        node_update_kernel<<<cdiv((size_t)N * HID, TB), TB, 0, stream>>>(
            h, agg, bgi, ht, N);
        inorm_kernel<<<NGRAPHS, HID, 0, stream>>>(ht, h, NP);
    }

    // 2) head: emb1 -> bn1 -> emb2 (h_out) -> bn2 @ w_pred (pred)
    gemm(h, w_emb1, b_emb1, z, N, HID, HID, 1);
    bn_kernel<<<cdiv((size_t)N * HID, TB), TB, 0, stream>>>(z, bn1_g, bn1_b,
                                                            bn1_m, bn1_v,
                                                            N * HID);
    gemm(z, w_emb2, b_emb2, out_h, N, HID, HID, 0);
    pred_kernel<<<cdiv(N, TB), TB, 0, stream>>>(out_h, bn2_g, bn2_b, bn2_m,
                                                bn2_v, w_pred, b_pred,
                                                out_pred, N);
}